// EarthMoverDistance_60241211293821
// MI455X (gfx1250) — compile-verified
//
#include <hip/hip_runtime.h>

typedef __attribute__((ext_vector_type(2))) float v2f;
typedef __attribute__((ext_vector_type(8))) float v8f;

#define BATCH 8
#define NPTS 2048
#define TILES (NPTS / 16)              // 128 column tiles per row strip
#define STRIPS (NPTS / 16)             // 128 row strips per batch
#define TOTAL_WAVES (BATCH * STRIPS)   // 1024 waves
#define WAVES_PER_BLOCK 8              // 256 threads (wave32)
#define AUCTION_EPS 0.005f
#define AUCTION_ITERS 50
#define NEG_INF (-3.402823466e38f)

// ---------------------------------------------------------------------------
// init: zero prices, inc, output accumulator
// ---------------------------------------------------------------------------
__global__ void emd_init_kernel(float* __restrict__ prices,
                                float* __restrict__ inc,
                                float* __restrict__ out) {
    int i = blockIdx.x * blockDim.x + threadIdx.x;
    if (i < BATCH * NPTS) {
        prices[i] = 0.0f;
        inc[i] = 0.0f;
    }
    if (i == 0) out[0] = 0.0f;
}

// ---------------------------------------------------------------------------
// bid: recompute S = 2*p.g - gn via WMMA f32 16x16x4, per-row top-2 of
// (S - prices), scatter-max bid into inc[best]. One wave per 16-row strip.
// All selection logic is branchless (v_cndmask / v_max / v_min).
// ---------------------------------------------------------------------------
__global__ __launch_bounds__(256) void emd_bid_kernel(
    const float* __restrict__ preds, const float* __restrict__ gts,
    const float* __restrict__ prices, float* __restrict__ inc) {
    __shared__ float s_price[NPTS];
    __shared__ float s_b[NPTS * 4];  // per col: gx, gy, gz, -|g|^2

    const int wave = threadIdx.x >> 5;
    const int lane = threadIdx.x & 31;
    const int gw = blockIdx.x * WAVES_PER_BLOCK + wave;
    const int batch = gw / STRIPS;          // uniform within a block
    const int strip = gw % STRIPS;

    const float* pb = preds + (size_t)batch * NPTS * 3;
    const float* gb = gts + (size_t)batch * NPTS * 3;

    for (int c = threadIdx.x; c < NPTS; c += 256) {
        float gx = gb[c * 3 + 0], gy = gb[c * 3 + 1], gz = gb[c * 3 + 2];
        s_b[c * 4 + 0] = gx;
        s_b[c * 4 + 1] = gy;
        s_b[c * 4 + 2] = gz;
        s_b[c * 4 + 3] = -(gx * gx + gy * gy + gz * gz);
        s_price[c] = prices[batch * NPTS + c];
    }
    __syncthreads();

    const int half = lane >> 4;  // 0: K0/K1, 1: K2/K3
    const int r = lane & 15;
    const int row = strip * 16 + r;

    float px = pb[row * 3 + 0], py = pb[row * 3 + 1], pz = pb[row * 3 + 2];
    v2f a;  // A 16x4 f32: lane<16 holds (K0,K1) for M=lane; lane>=16 holds (K2,K3)
    if (half == 0) { a.x = 2.0f * px; a.y = 2.0f * py; }
    else           { a.x = 2.0f * pz; a.y = 1.0f; }

    float t1[8], t2[8];
    int i1[8];
#pragma unroll
    for (int j = 0; j < 8; ++j) { t1[j] = NEG_INF; t2[j] = NEG_INF; i1[j] = 0; }

#pragma unroll 2
    for (int tile = 0; tile < TILES; ++tile) {
        const int col = tile * 16 + r;
        v2f b = *(const v2f*)&s_b[col * 4 + half * 2];  // B 4x16 mirrors A layout
        float price = s_price[col];
        v8f c = {};
        // D[m][n] = 2*p_m . g_n - |g_n|^2
        v8f d = __builtin_amdgcn_wmma_f32_16x16x4_f32(
            false, a, false, b, (short)0, c, false, false);
#pragma unroll
        for (int j = 0; j < 8; ++j) {   // D: N = lane%16, M = j + 8*half
            float v = d[j] - price;
            // branchless running top-2 (t2 <= t1 invariant; strict > keeps
            // the lowest column index on ties since cols ascend)
            t2[j] = fmaxf(t2[j], fminf(v, t1[j]));
            i1[j] = (v > t1[j]) ? col : i1[j];
            t1[j] = fmaxf(t1[j], v);
        }
    }

    // merge top-2 across the 16 lanes of each half (xor masks stay in-half)
#pragma unroll
    for (int mask = 1; mask < 16; mask <<= 1) {
#pragma unroll
        for (int j = 0; j < 8; ++j) {
            float o1 = __shfl_xor(t1[j], mask, 32);
            int oi1 = __shfl_xor(i1[j], mask, 32);
            float o2 = __shfl_xor(t2[j], mask, 32);
            bool takeO = (o1 > t1[j]) || (o1 == t1[j] && oi1 < i1[j]);
            float w2 = takeO ? o2 : t2[j];     // winner's top2
            float l1 = takeO ? t1[j] : o1;     // loser's top1
            t2[j] = fmaxf(l1, w2);
            t1[j] = takeO ? o1 : t1[j];
            i1[j] = takeO ? oi1 : i1[j];
        }
    }

    if (r == 0) {  // lanes 0 and 16 own rows strip*16 + j + 8*half
#pragma unroll
        for (int j = 0; j < 8; ++j) {
            float bid = t1[j] - t2[j] + AUCTION_EPS;  // > 0
            atomicMax((int*)&inc[batch * NPTS + i1[j]], __float_as_int(bid));
        }
    }
}

// ---------------------------------------------------------------------------
// update: prices += inc; inc = 0
// ---------------------------------------------------------------------------
__global__ void emd_update_kernel(float* __restrict__ prices,
                                  float* __restrict__ inc) {
    int i = blockIdx.x * blockDim.x + threadIdx.x;
    if (i < BATCH * NPTS) {
        prices[i] += inc[i];
        inc[i] = 0.0f;
    }
}

// ---------------------------------------------------------------------------
// final: per-row argmax of (S - prices), accumulate matched ||p-g||^2 per wave
// ---------------------------------------------------------------------------
__global__ __launch_bounds__(256) void emd_final_kernel(
    const float* __restrict__ preds, const float* __restrict__ gts,
    const float* __restrict__ prices, float* __restrict__ partials) {
    __shared__ float s_price[NPTS];
    __shared__ float s_b[NPTS * 4];

    const int wave = threadIdx.x >> 5;
    const int lane = threadIdx.x & 31;
    const int gw = blockIdx.x * WAVES_PER_BLOCK + wave;
    const int batch = gw / STRIPS;
    const int strip = gw % STRIPS;

    const float* pb = preds + (size_t)batch * NPTS * 3;
    const float* gb = gts + (size_t)batch * NPTS * 3;

    for (int c = threadIdx.x; c < NPTS; c += 256) {
        float gx = gb[c * 3 + 0], gy = gb[c * 3 + 1], gz = gb[c * 3 + 2];
        s_b[c * 4 + 0] = gx;
        s_b[c * 4 + 1] = gy;
        s_b[c * 4 + 2] = gz;
        s_b[c * 4 + 3] = -(gx * gx + gy * gy + gz * gz);
        s_price[c] = prices[batch * NPTS + c];
    }
    __syncthreads();

    const int half = lane >> 4;
    const int r = lane & 15;
    const int row = strip * 16 + r;

    float px = pb[row * 3 + 0], py = pb[row * 3 + 1], pz = pb[row * 3 + 2];
    v2f a;
    if (half == 0) { a.x = 2.0f * px; a.y = 2.0f * py; }
    else           { a.x = 2.0f * pz; a.y = 1.0f; }

    float t1[8];
    int i1[8];
#pragma unroll
    for (int j = 0; j < 8; ++j) { t1[j] = NEG_INF; i1[j] = 0; }

#pragma unroll 2
    for (int tile = 0; tile < TILES; ++tile) {
        const int col = tile * 16 + r;
        v2f b = *(const v2f*)&s_b[col * 4 + half * 2];
        float price = s_price[col];
        v8f c = {};
        v8f d = __builtin_amdgcn_wmma_f32_16x16x4_f32(
            false, a, false, b, (short)0, c, false, false);
#pragma unroll
        for (int j = 0; j < 8; ++j) {
            float v = d[j] - price;
            i1[j] = (v > t1[j]) ? col : i1[j];   // branchless argmax
            t1[j] = fmaxf(t1[j], v);
        }
    }

#pragma unroll
    for (int mask = 1; mask < 16; mask <<= 1) {
#pragma unroll
        for (int j = 0; j < 8; ++j) {
            float o1 = __shfl_xor(t1[j], mask, 32);
            int oi1 = __shfl_xor(i1[j], mask, 32);
            bool takeO = (o1 > t1[j]) || (o1 == t1[j] && oi1 < i1[j]);
            t1[j] = takeO ? o1 : t1[j];
            i1[j] = takeO ? oi1 : i1[j];
        }
    }

    float acc = 0.0f;
    if (r == 0) {
#pragma unroll
        for (int j = 0; j < 8; ++j) {
            int rr = strip * 16 + j + 8 * half;
            float qx = pb[rr * 3 + 0], qy = pb[rr * 3 + 1], qz = pb[rr * 3 + 2];
            int best = i1[j];
            float dx = qx - s_b[best * 4 + 0];
            float dy = qy - s_b[best * 4 + 1];
            float dz = qz - s_b[best * 4 + 2];
            acc += dx * dx + dy * dy + dz * dz;  // matched squared distance
        }
    }
    acc += __shfl_xor(acc, 16, 32);  // combine the two halves' owners
    if (lane == 0) partials[gw] = acc;
}

// ---------------------------------------------------------------------------
// reduce: one wave sums the 1024 per-strip partials deterministically
// ---------------------------------------------------------------------------
__global__ void emd_reduce_kernel(const float* __restrict__ partials,
                                  float* __restrict__ out) {
    int lane = threadIdx.x & 31;
    float s = 0.0f;
    for (int i = lane; i < TOTAL_WAVES; i += 32) s += partials[i];
#pragma unroll
    for (int mask = 16; mask >= 1; mask >>= 1) s += __shfl_xor(s, mask, 32);
    if (lane == 0) out[0] = s;
}

extern "C" void kernel_launch(void* const* d_in, const int* in_sizes, int n_in,
                              void* d_out, int out_size, void* d_ws, size_t ws_size,
                              hipStream_t stream) {
    (void)in_sizes; (void)n_in; (void)out_size; (void)ws_size;
    const float* preds = (const float*)d_in[0];
    const float* gts = (const float*)d_in[1];
    float* out = (float*)d_out;

    float* prices = (float*)d_ws;                // BATCH*NPTS floats
    float* inc = prices + BATCH * NPTS;          // BATCH*NPTS floats
    float* partials = inc + BATCH * NPTS;        // TOTAL_WAVES floats

    const int grid_elem = (BATCH * NPTS + 255) / 256;
    const int grid_wave = TOTAL_WAVES / WAVES_PER_BLOCK;

    emd_init_kernel<<<grid_elem, 256, 0, stream>>>(prices, inc, out);
    for (int it = 0; it < AUCTION_ITERS; ++it) {
        emd_bid_kernel<<<grid_wave, 256, 0, stream>>>(preds, gts, prices, inc);
        emd_update_kernel<<<grid_elem, 256, 0, stream>>>(prices, inc);
    }
    emd_final_kernel<<<grid_wave, 256, 0, stream>>>(preds, gts, prices, partials);
    emd_reduce_kernel<<<1, 32, 0, stream>>>(partials, out);
}